// sub_sumgnn_28638841930489
// MI455X (gfx1250) — compile-verified
//
#include <hip/hip_runtime.h>
#include <hip/hip_bf16.h>
#include <math.h>

typedef __attribute__((ext_vector_type(2))) float v2f;
typedef __attribute__((ext_vector_type(8))) float v8f;

#define GN   100000   // nodes
#define GE   800000   // edges per sign
#define GB   4096     // idx batch

// LDS W-chunk layout: row-pair interleaved. Element (k, col) lives at
//   (k>>1)*LDSP + col*2 + (k&1)
// so {W[kr][col], W[kr+1][col]} is one aligned 64-bit DS load.
// LDSP = 2*64 + 32: pair-blocks differ by 32 banks -> the two half-waves
// (rows kr vs kr+2) hit disjoint bank halves: conflict-free ds_load_b64.
#define LDSP 160

// ---------------------------------------------------------------------------
// WMMA fp32 GEMM: C[M,64] = act(A[M,K] @ W[K,64] + bias), K % 64 == 0
// block = 256 threads = 8 waves; each wave owns 16 rows x 64 cols (4 wmma
// tiles sharing one A fragment); block covers 128 rows.
// ACT: 0=none, 1=tanh, 2=relu
// ---------------------------------------------------------------------------
template <int ACT>
__global__ __launch_bounds__(256) void gemm64_wmma(
    const float* __restrict__ A, int lda,
    const float* __restrict__ W,
    const float* __restrict__ bias,
    float* __restrict__ C, int ldc,
    int M, int K)
{
    __shared__ float ldsW[32 * LDSP];   // 20 KB: one 64x64 W chunk

    const int wave = threadIdx.x >> 5;
    const int lane = threadIdx.x & 31;
    const int half = lane >> 4;       // 0: lanes 0-15, 1: lanes 16-31
    const int l16  = lane & 15;

    const int rowTile = blockIdx.x * 128 + wave * 16;
    int aRow = rowTile + l16;
    if (aRow >= M) aRow = M - 1;      // branchless clamp keeps EXEC all-ones

    v8f acc[4] = {};

    for (int k0 = 0; k0 < K; k0 += 64) {
        __syncthreads();
        // cooperative load of 64x64 W chunk into pair-interleaved layout
        for (int i = threadIdx.x; i < 64 * 64; i += 256) {
            const int k = i >> 6, col = i & 63;
            ldsW[(k >> 1) * LDSP + col * 2 + (k & 1)] =
                W[(size_t)(k0 + k) * 64 + col];
        }
        // prefetch next chunk while this one is consumed
        if (k0 + 64 < K)
            __builtin_prefetch(&W[(size_t)(k0 + 64 + (threadIdx.x >> 2)) * 64], 0, 0);
        __syncthreads();

        const float* ap = A + (size_t)aRow * lda + k0 + half * 2;
        #pragma unroll 4
        for (int k = 0; k < 64; k += 4) {
            // A 16x4 fp32 tile: lanes 0-15 hold K=k,k+1 ; lanes 16-31 hold K=k+2,k+3
            v2f a;
            a.x = ap[k + 0];
            a.y = ap[k + 1];
            // B 4x16 fp32 tiles (cols 0,16,32,48): one ds_load_b64 each
            const int kr = k + half * 2;
            const float* wp = &ldsW[(kr >> 1) * LDSP + l16 * 2];
            #pragma unroll
            for (int t = 0; t < 4; ++t) {
                const v2f b = *(const v2f*)(wp + t * 32);
                acc[t] = __builtin_amdgcn_wmma_f32_16x16x4_f32(
                    false, a, false, b, (short)0, acc[t], false, false);
            }
        }
    }

    // D layout: VGPR r -> row rowTile + half*8 + r, col tile*16 + l16
    const int orow = rowTile + half * 8;
    #pragma unroll
    for (int t = 0; t < 4; ++t) {
        const int ocol = t * 16 + l16;
        const float bb = bias[ocol];
        #pragma unroll
        for (int r = 0; r < 8; ++r) {
            const int rr = orow + r;
            if (rr < M) {
                float v = acc[t][r] + bb;
                if (ACT == 1) v = tanhf(v);
                else if (ACT == 2) v = fmaxf(v, 0.0f);
                C[(size_t)rr * ldc + ocol] = v;
            }
        }
    }
}

// ---------------------------------------------------------------------------
// Per-node attention scalars: s[j][n] = h[n,:] . wv[j], j = 2k+half, k=0..3
// attw slice for a layer is [4][128] contiguous == wv[8][64] contiguous.
// ---------------------------------------------------------------------------
__global__ __launch_bounds__(256) void node_scores(
    const float* __restrict__ h, const float* __restrict__ attw,
    int n_nodes, float* __restrict__ s)
{
    __shared__ float wv[8 * 64];
    for (int i = threadIdx.x; i < 512; i += 256) wv[i] = attw[i];
    __syncthreads();

    const int n = blockIdx.x * 256 + threadIdx.x;
    if (n >= n_nodes) return;

    const float* hp = h + (size_t)n * 64;
    float acc[8] = {0, 0, 0, 0, 0, 0, 0, 0};
    for (int f = 0; f < 64; ++f) {
        const float hv = hp[f];
        #pragma unroll
        for (int j = 0; j < 8; ++j) acc[j] += hv * wv[j * 64 + f];
    }
    #pragma unroll
    for (int j = 0; j < 8; ++j) s[(size_t)j * n_nodes + n] = acc[j];
}

// ---------------------------------------------------------------------------
// Zero buffer (graph-capture-safe; float4 stores)
// ---------------------------------------------------------------------------
__global__ __launch_bounds__(256) void zero4(float4* __restrict__ p, size_t n4)
{
    size_t i = (size_t)blockIdx.x * 256 + threadIdx.x;
    if (i < n4) p[i] = make_float4(0.f, 0.f, 0.f, 0.f);
}

// ---------------------------------------------------------------------------
// Edge scatter: one wave per edge, 2 features per lane, both directions of
// one edge-sign. out4 is [N, 256], direction k writes column block k*64.
//   dir1 (k=kbase):   src=row dst=col  score=exp(tanh(s1k[row]+s2k[col]+b_k))
//   dir2 (k=kbase+1): src=col dst=row
// ---------------------------------------------------------------------------
__global__ __launch_bounds__(256) void edge_scatter(
    const int* __restrict__ rows, const int* __restrict__ cols, int n_edges,
    const float* __restrict__ h, const float* __restrict__ s, int n_nodes,
    const float* __restrict__ att_b, int kbase,
    float* __restrict__ out4)
{
    const int lane = threadIdx.x & 31;
    const int e = blockIdx.x * 8 + (threadIdx.x >> 5);
    if (e >= n_edges) return;

    const int r = rows[e];
    const int c = cols[e];
    const int k0 = kbase, k1 = kbase + 1;

    // uniform per-wave scalars (same-address loads broadcast from cache)
    const float sc0 = expf(tanhf(s[(size_t)(2 * k0)     * n_nodes + r] +
                                 s[(size_t)(2 * k0 + 1) * n_nodes + c] + att_b[k0]));
    const float sc1 = expf(tanhf(s[(size_t)(2 * k1)     * n_nodes + c] +
                                 s[(size_t)(2 * k1 + 1) * n_nodes + r] + att_b[k1]));

    const float2 vc = ((const float2*)(h + (size_t)c * 64))[lane];
    const float2 vr = ((const float2*)(h + (size_t)r * 64))[lane];

    float* o0 = out4 + (size_t)r * 256 + k0 * 64 + lane * 2;
    atomicAdd(o0,     vc.x * sc0);
    atomicAdd(o0 + 1, vc.y * sc0);
    float* o1 = out4 + (size_t)c * 256 + k1 * 64 + lane * 2;
    atomicAdd(o1,     vr.x * sc1);
    atomicAdd(o1 + 1, vr.y * sc1);
}

// ---------------------------------------------------------------------------
// x12[b] = cat(zc[idx1[b]], zc[idx2[b]]) ; x21[b] = cat(zc[idx2[b]], zc[idx1[b]])
// one thread per (b, j<128)
// ---------------------------------------------------------------------------
__global__ __launch_bounds__(256) void gather_pairs(
    const float* __restrict__ zc, const int* __restrict__ i1,
    const int* __restrict__ i2, float* __restrict__ x12,
    float* __restrict__ x21, int nb)
{
    const int t = blockIdx.x * 256 + threadIdx.x;
    if (t >= nb * 128) return;
    const int b = t >> 7, j = t & 127;
    const float a = zc[(size_t)i1[b] * 128 + j];
    const float c = zc[(size_t)i2[b] * 128 + j];
    x12[(size_t)b * 256 + j]       = a;
    x12[(size_t)b * 256 + 128 + j] = c;
    x21[(size_t)b * 256 + j]       = c;
    x21[(size_t)b * 256 + 128 + j] = a;
}

// ---------------------------------------------------------------------------
// out[b, 0:2] = Hin[b,:] @ W[64,2] + bias
// ---------------------------------------------------------------------------
__global__ __launch_bounds__(256) void head2(
    const float* __restrict__ Hin, const float* __restrict__ W,
    const float* __restrict__ bias, float* __restrict__ out, int M)
{
    const int i = blockIdx.x * 256 + threadIdx.x;
    if (i >= M) return;
    const float* hp = Hin + (size_t)i * 64;
    float a0 = bias[0], a1 = bias[1];
    for (int f = 0; f < 64; ++f) {
        const float v = hp[f];
        a0 += v * W[f * 2 + 0];
        a1 += v * W[f * 2 + 1];
    }
    out[(size_t)i * 2 + 0] = a0;
    out[(size_t)i * 2 + 1] = a1;
}

// ---------------------------------------------------------------------------
extern "C" void kernel_launch(void* const* d_in, const int* in_sizes, int n_in,
                              void* d_out, int out_size, void* d_ws, size_t ws_size,
                              hipStream_t stream)
{
    (void)in_sizes; (void)n_in; (void)out_size; (void)ws_size;

    const float* x       = (const float*)d_in[0];
    const int*   ei      = (const int*)  d_in[1];   // [2,E]
    const int*   ein     = (const int*)  d_in[2];   // [2,E]
    const int*   idx1    = (const int*)  d_in[3];
    const int*   idx2    = (const int*)  d_in[4];
    const float* lin_w0  = (const float*)d_in[5];
    const float* lin_b0  = (const float*)d_in[6];
    const float* lin_w1  = (const float*)d_in[7];
    const float* lin_b1  = (const float*)d_in[8];
    const float* att_w   = (const float*)d_in[9];   // [L,4,128]
    const float* att_b   = (const float*)d_in[10];  // [L,4]
    const float* cat_w   = (const float*)d_in[11];  // [L,256,64]
    const float* cat_b   = (const float*)d_in[12];  // [L,64]
    const float* sw1 = (const float*)d_in[13]; const float* sb1 = (const float*)d_in[14];
    const float* sw2 = (const float*)d_in[15]; const float* sb2 = (const float*)d_in[16];
    const float* sw3 = (const float*)d_in[17]; const float* sb3 = (const float*)d_in[18];
    const float* sw4 = (const float*)d_in[19]; const float* sb4 = (const float*)d_in[20];
    const float* dw1 = (const float*)d_in[21]; const float* db1 = (const float*)d_in[22];
    const float* dw2 = (const float*)d_in[23]; const float* db2 = (const float*)d_in[24];

    float* outp = (float*)d_out;

    // workspace layout (floats)
    float* ws   = (float*)d_ws;
    float* h    = ws;                          // N*64
    float* out4 = h    + (size_t)GN * 64;      // N*256
    float* zc   = out4 + (size_t)GN * 256;     // N*128
    float* s    = zc   + (size_t)GN * 128;     // 8*N
    float* x12  = s    + (size_t)8 * GN;       // B*256
    float* x21  = x12  + (size_t)GB * 256;     // B*256
    float* t0   = x21  + (size_t)GB * 256;     // B*64
    float* t1   = t0   + (size_t)GB * 64;      // B*64

    const float* lw[2] = {lin_w0, lin_w1};
    const float* lb[2] = {lin_b0, lin_b1};

    const int gridN = (GN + 127) / 128;
    const int gridB = (GB + 127) / 128;

    const float* zin = x;
    int lda = 64;
    for (int l = 0; l < 2; ++l) {
        // h = zin @ lin_w[l] + lin_b[l]
        gemm64_wmma<0><<<gridN, 256, 0, stream>>>(zin, lda, lw[l], lb[l], h, 64, GN, 64);
        // attention scalars for the 4 combos
        node_scores<<<(GN + 255) / 256, 256, 0, stream>>>(h, att_w + (size_t)l * 512, GN, s);
        // out4 = 0
        {
            const size_t n4 = (size_t)GN * 256 / 4;
            zero4<<<(unsigned)((n4 + 255) / 256), 256, 0, stream>>>((float4*)out4, n4);
        }
        // scatter pos edges (combos 0,1) and neg edges (combos 2,3)
        edge_scatter<<<(GE + 7) / 8, 256, 0, stream>>>(ei,  ei  + GE, GE, h, s, GN, att_b + l * 4, 0, out4);
        edge_scatter<<<(GE + 7) / 8, 256, 0, stream>>>(ein, ein + GE, GE, h, s, GN, att_b + l * 4, 2, out4);
        // z = tanh(out4 @ cat_w[l] + cat_b[l]) written into zc[:, l*64 : l*64+64]
        gemm64_wmma<1><<<gridN, 256, 0, stream>>>(out4, 256,
                                                  cat_w + (size_t)l * 256 * 64,
                                                  cat_b + (size_t)l * 64,
                                                  zc + (size_t)l * 64, 128, GN, 256);
        zin = zc;      // next layer reads columns [0,64) of zc
        lda = 128;
    }

    // pair gathering
    gather_pairs<<<(GB * 128 + 255) / 256, 256, 0, stream>>>(zc, idx1, idx2, x12, x21, GB);

    // sign head: relu MLP 256->64->64->64 then 64->2
    gemm64_wmma<2><<<gridB, 256, 0, stream>>>(x12, 256, sw1, sb1, t0, 64, GB, 256);
    gemm64_wmma<2><<<gridB, 256, 0, stream>>>(t0,  64,  sw2, sb2, t1, 64, GB, 64);
    gemm64_wmma<2><<<gridB, 256, 0, stream>>>(t1,  64,  sw3, sb3, t0, 64, GB, 64);
    head2<<<(GB + 255) / 256, 256, 0, stream>>>(t0, sw4, sb4, outp, GB);

    // direction head: relu(v@dw1+db1)@dw2+db2 for x12 then x21
    gemm64_wmma<2><<<gridB, 256, 0, stream>>>(x12, 256, dw1, db1, t1, 64, GB, 256);
    head2<<<(GB + 255) / 256, 256, 0, stream>>>(t1, dw2, db2, outp + 2 * GB, GB);
    gemm64_wmma<2><<<gridB, 256, 0, stream>>>(x21, 256, dw1, db1, t1, 64, GB, 256);
    head2<<<(GB + 255) / 256, 256, 0, stream>>>(t1, dw2, db2, outp + 4 * GB, GB);
}